// SkipGNN_31258771980721
// MI455X (gfx1250) — compile-verified
//
#include <hip/hip_runtime.h>

// ---------------------------------------------------------------------------
// SkipGNN on MI455X (gfx1250, wave32).
// Roofline: SpMM scatter traffic (~10 GB) bounds runtime at HBM/atomic rate;
// dense GEMMs (~31 GFLOP f32) run on the WMMA pipe (V_WMMA_F32_16X16X4_F32),
// exact f32 math matching the reference.
// Fast GEMM path: one wave = 16x64 output strip, 4 WMMAs per A-fragment load,
// no predication (Nout % 64 == 0). Tail path handles Nout=86 with clamped
// loads + value selects (no exec juggling).
// Workspace requirement: 4 * N * 128 * 4 bytes (~205 MB) in d_ws.
// ---------------------------------------------------------------------------

#define NFEAT 256
#define NHID  128
#define RELL  86

typedef float v2f __attribute__((ext_vector_type(2)));
typedef float v8f __attribute__((ext_vector_type(8)));

#define WMMA_F32(a, b, c)                                             \
  __builtin_amdgcn_wmma_f32_16x16x4_f32(false, (a), false, (b),       \
                                        (short)0, (c), false, false)

// ---- Fast path: C[M x Nout] = A[M x K] @ W[K x Nout] (+bias), Nout % 64 == 0.
// blockDim = (32,4). Wave = 16 rows x 64 cols (4 col-tiles, 4 accumulators).
// A-frag (16x4 f32): lanes 0-15 hold K=k+0,k+1 ; lanes 16-31 hold K=k+2,k+3.
// B-frag (4x16 f32): symmetric, N indexed by lane%16.
// C/D (16x16 f32, 8 VGPRs): VGPR r -> M = r + 8*(lane/16), N = lane%16.
__global__ __launch_bounds__(128)
void gemm_wmma_f32_n64(const float* __restrict__ A, const float* __restrict__ W,
                       const float* __restrict__ bias, float* __restrict__ C,
                       int M, int K, int Nout) {
  const int lane = threadIdx.x;        // 0..31
  const int half = lane >> 4;          // 0 or 1
  const int m    = lane & 15;
  const int row0 = (blockIdx.y * 4 + threadIdx.y) * 16;
  if (row0 >= M) return;               // wave-uniform exit (EXEC stays all-1s)
  const int col  = blockIdx.x * 64 + m;

  const float* __restrict__ Arow = A + (size_t)(row0 + m) * K;
  const float* __restrict__ Wc   = W + col;

  v8f acc0 = {}, acc1 = {}, acc2 = {}, acc3 = {};
  for (int k = 0; k < K; k += 4) {
    const int ka = k + 2 * half;
    v2f a;
    a.x = Arow[ka];
    a.y = Arow[ka + 1];
    const float* __restrict__ w0 = Wc + (size_t)ka * Nout;
    const float* __restrict__ w1 = Wc + (size_t)(ka + 1) * Nout;
    v2f b0, b1, b2, b3;
    b0.x = w0[0];  b0.y = w1[0];
    b1.x = w0[16]; b1.y = w1[16];
    b2.x = w0[32]; b2.y = w1[32];
    b3.x = w0[48]; b3.y = w1[48];
    acc0 = WMMA_F32(a, b0, acc0);
    acc1 = WMMA_F32(a, b1, acc1);
    acc2 = WMMA_F32(a, b2, acc2);
    acc3 = WMMA_F32(a, b3, acc3);
  }

  const float bv0 = bias ? bias[col]      : 0.0f;
  const float bv1 = bias ? bias[col + 16] : 0.0f;
  const float bv2 = bias ? bias[col + 32] : 0.0f;
  const float bv3 = bias ? bias[col + 48] : 0.0f;
#pragma unroll
  for (int r = 0; r < 8; ++r) {
    float* __restrict__ Crow = C + (size_t)(row0 + r + 8 * half) * Nout + col;
    Crow[0]  = acc0[r] + bv0;
    Crow[16] = acc1[r] + bv1;
    Crow[32] = acc2[r] + bv2;
    Crow[48] = acc3[r] + bv3;
  }
}

// ---- Tail path: one 16x16 tile per wave, ragged Nout (dec2: 86).
// Out-of-range cols: clamp the load address, zero the value via cndmask.
__global__ __launch_bounds__(128)
void gemm_wmma_f32_tail(const float* __restrict__ A, const float* __restrict__ W,
                        const float* __restrict__ bias, float* __restrict__ C,
                        int M, int K, int Nout) {
  const int lane = threadIdx.x;
  const int half = lane >> 4;
  const int m    = lane & 15;
  const int row0 = blockIdx.y * 16;
  const int col0 = (blockIdx.x * 4 + threadIdx.y) * 16;
  if (row0 >= M || col0 >= Nout) return;      // wave-uniform
  const int  col   = col0 + m;
  const bool colOK = (col < Nout);
  const int  colc  = colOK ? col : 0;         // clamped address, always valid

  const float* __restrict__ Arow = A + (size_t)(row0 + m) * K;
  v8f acc = {};
  for (int k = 0; k < K; k += 4) {
    const int ka = k + 2 * half;
    v2f a, b;
    a.x = Arow[ka];
    a.y = Arow[ka + 1];
    float w0 = W[(size_t)ka * Nout + colc];        // unconditional loads
    float w1 = W[(size_t)(ka + 1) * Nout + colc];
    b.x = colOK ? w0 : 0.0f;                       // value select, not exec
    b.y = colOK ? w1 : 0.0f;
    acc = WMMA_F32(a, b, acc);
  }

  if (colOK) {
    const float bv = bias ? bias[col] : 0.0f;
#pragma unroll
    for (int r = 0; r < 8; ++r)
      C[(size_t)(row0 + r + 8 * half) * Nout + col] = acc[r] + bv;
  }
}

// COO SpMM scatter: out[row,:] += val * S[col,:]. One edge per 128-thread block
// iteration; feature gather is a coalesced 512B row; scatter via HW f32 atomic.
__global__ __launch_bounds__(128)
void spmm_coo_atomic(const int* __restrict__ rows, const int* __restrict__ cols,
                     const float* __restrict__ vals, const float* __restrict__ S,
                     float* __restrict__ out, int E) {
  const int f = threadIdx.x;           // 0..127 feature lane
  for (int e = blockIdx.x; e < E; e += gridDim.x) {
    const int   r = rows[e];
    const int   c = cols[e];
    const float v = vals[e];
    const float s = S[(size_t)c * NHID + f];
    unsafeAtomicAdd(&out[(size_t)r * NHID + f], v * s);
  }
}

__global__ __launch_bounds__(256)
void fill_zero(float* __restrict__ p, size_t n) {
  size_t i      = (size_t)blockIdx.x * blockDim.x + threadIdx.x;
  size_t stride = (size_t)gridDim.x * blockDim.x;
  for (; i < n; i += stride) p[i] = 0.0f;
}

// p[i] += b1[col] + b2[col]; optional ReLU.  Feature width fixed at NHID=128.
__global__ __launch_bounds__(256)
void bias_act(float* __restrict__ p, const float* __restrict__ b1,
              const float* __restrict__ b2, size_t total, int doRelu) {
  size_t i      = (size_t)blockIdx.x * blockDim.x + threadIdx.x;
  size_t stride = (size_t)gridDim.x * blockDim.x;
  for (; i < total; i += stride) {
    const int col = (int)(i & (NHID - 1));
    float v = p[i] + b1[col] + b2[col];
    p[i] = doRelu ? fmaxf(v, 0.0f) : v;
  }
}

// feat[b] = concat(h[idx0[b]], h[idx1[b]]) ; idx flat [2,B].
__global__ __launch_bounds__(256)
void gather_concat(const float* __restrict__ h, const int* __restrict__ idx,
                   float* __restrict__ feat, int B) {
  const int b = blockIdx.x;
  const int t = threadIdx.x;           // 0..255
  const int src = (t < NHID) ? idx[b] : idx[B + b];
  feat[(size_t)b * (2 * NHID) + t] = h[(size_t)src * NHID + (t & (NHID - 1))];
}

extern "C" void kernel_launch(void* const* d_in, const int* in_sizes, int n_in,
                              void* d_out, int out_size, void* d_ws, size_t ws_size,
                              hipStream_t stream) {
  const float* x       = (const float*)d_in[0];
  const int*   o_rows  = (const int*)  d_in[1];
  const int*   o_cols  = (const int*)  d_in[2];
  const float* o_vals  = (const float*)d_in[3];
  const int*   s_rows  = (const int*)  d_in[4];
  const int*   s_cols  = (const int*)  d_in[5];
  const float* s_vals  = (const float*)d_in[6];
  const int*   idx     = (const int*)  d_in[7];
  const float* W_ogc1  = (const float*)d_in[8];
  const float* b_ogc1  = (const float*)d_in[9];
  const float* W_sgc1o = (const float*)d_in[10];
  const float* b_sgc1o = (const float*)d_in[11];
  const float* W_sgc1  = (const float*)d_in[12];
  const float* b_sgc1  = (const float*)d_in[13];
  const float* W_ogc1s = (const float*)d_in[14];
  const float* b_ogc1s = (const float*)d_in[15];
  const float* W_ogc2  = (const float*)d_in[16];
  const float* b_ogc2  = (const float*)d_in[17];
  const float* W_sgc2o = (const float*)d_in[18];
  const float* b_sgc2o = (const float*)d_in[19];
  const float* W_dec1  = (const float*)d_in[20];
  const float* b_dec1  = (const float*)d_in[21];
  const float* W_dec2  = (const float*)d_in[22];
  const float* b_dec2  = (const float*)d_in[23];

  const int N = in_sizes[0] / NFEAT;   // 100000
  const int E = in_sizes[1];           // 1600000
  const int B = in_sizes[7] / 2;       // 16384

  const size_t NB = (size_t)N * NHID;  // elements per node buffer
  float* bufA = (float*)d_ws;          // scratch buffers (4 x ~51.2 MB)
  float* bufB = bufA + NB;
  float* bufC = bufB + NB;
  float* bufD = bufC + NB;

  const dim3 gemmBlk(32, 4);
  auto launch_gemm = [&](const float* Aa, const float* Ww, const float* bb,
                         float* Cc, int M, int K, int Nout) {
    if ((Nout & 63) == 0) {
      dim3 grid((unsigned)(Nout / 64), (unsigned)((M / 16 + 3) / 4));
      gemm_wmma_f32_n64<<<grid, gemmBlk, 0, stream>>>(Aa, Ww, bb, Cc, M, K, Nout);
    } else {
      dim3 grid((unsigned)(((Nout + 15) / 16 + 3) / 4), (unsigned)(M / 16));
      gemm_wmma_f32_tail<<<grid, gemmBlk, 0, stream>>>(Aa, Ww, bb, Cc, M, K, Nout);
    }
  };
  auto launch_zero = [&](float* p, size_t n) {
    fill_zero<<<2048, 256, 0, stream>>>(p, n);
  };
  auto launch_spmm = [&](const int* r, const int* c, const float* v,
                         const float* S, float* outp) {
    spmm_coo_atomic<<<32768, 128, 0, stream>>>(r, c, v, S, outp, E);
  };
  auto launch_bias = [&](float* p, const float* b1, const float* b2, int relu) {
    bias_act<<<4096, 256, 0, stream>>>(p, b1, b2, NB, relu);
  };

  // ---- Layer 1: o_x = relu(spmm(o, x@W_ogc1) + spmm(s, x@W_sgc1o) + biases)
  launch_gemm(x, W_ogc1, nullptr, bufA, N, NFEAT, NHID);
  launch_gemm(x, W_sgc1o, nullptr, bufB, N, NFEAT, NHID);
  launch_zero(bufC, NB);
  launch_spmm(o_rows, o_cols, o_vals, bufA, bufC);
  launch_spmm(s_rows, s_cols, s_vals, bufB, bufC);
  launch_bias(bufC, b_ogc1, b_sgc1o, 1);                 // bufC = o_x

  // ---- Layer 1b: s_x = relu(spmm(s, x@W_sgc1) + spmm(o, o_x@W_ogc1s) + biases)
  launch_gemm(x, W_sgc1, nullptr, bufA, N, NFEAT, NHID);
  launch_gemm(bufC, W_ogc1s, nullptr, bufB, N, NHID, NHID);
  launch_zero(bufD, NB);
  launch_spmm(s_rows, s_cols, s_vals, bufA, bufD);
  launch_spmm(o_rows, o_cols, o_vals, bufB, bufD);
  launch_bias(bufD, b_sgc1, b_ogc1s, 1);                 // bufD = s_x

  // ---- Layer 2: h = spmm(o, o_x@W_ogc2) + spmm(s, s_x@W_sgc2o) + biases
  launch_gemm(bufC, W_ogc2, nullptr, bufA, N, NHID, NHID);
  launch_gemm(bufD, W_sgc2o, nullptr, bufB, N, NHID, NHID);
  launch_zero(bufC, NB);                                 // o_x no longer needed
  launch_spmm(o_rows, o_cols, o_vals, bufA, bufC);
  launch_spmm(s_rows, s_cols, s_vals, bufB, bufC);
  launch_bias(bufC, b_ogc2, b_sgc2o, 0);                 // bufC = h

  // ---- Decoder: out = (concat(h[idx0], h[idx1]) @ W_dec1 + b1) @ W_dec2 + b2
  gather_concat<<<B, 256, 0, stream>>>(bufC, idx, bufA, B);   // bufA = feat [B,256]
  launch_gemm(bufA, W_dec1, b_dec1, bufB, B, 2 * NHID, NHID); // bufB = d1 [B,128]
  launch_gemm(bufB, W_dec2, b_dec2, (float*)d_out, B, NHID, RELL);
}